// TripletLoss_1125281432196
// MI455X (gfx1250) — compile-verified
//
#include <hip/hip_runtime.h>
#include <math.h>

typedef __attribute__((ext_vector_type(2))) float v2f;
typedef __attribute__((ext_vector_type(8))) float v8f;

#define NROWS 8192
#define KDIM  128
#define TILE  16
#define NTILES (NROWS / TILE)      // 512
#define WAVES_PER_BLOCK 8
#define EPSF 1e-12f

// ---------------------------------------------------------------------------
// Kernel A: squared row norms  sq[i] = ||x_i||^2
// ---------------------------------------------------------------------------
__global__ void sqnorm_kernel(const float* __restrict__ x, float* __restrict__ sq) {
    int row = blockIdx.x * blockDim.x + threadIdx.x;
    if (row >= NROWS) return;
    const float4* p = reinterpret_cast<const float4*>(x + (size_t)row * KDIM);
    float s = 0.0f;
#pragma unroll
    for (int i = 0; i < KDIM / 4; ++i) {
        float4 v = p[i];
        s += v.x * v.x + v.y * v.y + v.z * v.z + v.w * v.w;
    }
    sq[row] = s;
}

// ---------------------------------------------------------------------------
// Kernel B: fused Gram (fp32 WMMA 16x16x4) + batch-hard mining.
// One block per 16-row tile; 8 waves split the 512 column tiles, processing
// TWO column tiles per iteration (4 independent WMMA chains, shared A regs).
// Per-lane operand addressing for V_WMMA_F32_16X16X4_F32:
//   A[lane][v] = X[i0 + (lane&15)][4*kk + 2*(lane>>4) + v]
//   B[lane][v] = X[j0 + (lane&15)][4*kk + 2*(lane>>4) + v]
//   acc[v] @ lane = C[m = v + 8*(lane>>4)][n = lane&15]
// ---------------------------------------------------------------------------
__global__ __launch_bounds__(256) void gram_hard_kernel(
    const float* __restrict__ x, const long long* __restrict__ tgt,
    const float* __restrict__ sq, const float* __restrict__ marginp,
    float* __restrict__ rowloss) {

    const int lane = threadIdx.x & 31;
    const int wave = threadIdx.x >> 5;
    const int half = lane >> 4;      // 0 or 1
    const int lm   = lane & 15;
    const int i0   = blockIdx.x * TILE;

    // A-tile resident in registers for the whole kernel (64 VGPRs/lane).
    const float* arow = x + (size_t)(i0 + lm) * KDIM + 2 * half;
    v2f a[KDIM / 4];
#pragma unroll
    for (int kk = 0; kk < KDIM / 4; ++kk)
        a[kk] = *reinterpret_cast<const v2f*>(arow + 4 * kk);

    float sqi[8];
    int   ti[8];
#pragma unroll
    for (int v = 0; v < 8; ++v) {
        int r  = i0 + v + 8 * half;
        sqi[v] = sq[r];
        ti[v]  = (int)tgt[r];        // labels are 0..31, low word suffices
    }

    float ap[8], an[8];
#pragma unroll
    for (int v = 0; v < 8; ++v) { ap[v] = -INFINITY; an[v] = INFINITY; }

    // Each wave covers column-tile residues {wave, wave+8} (mod 16).
    for (int jt = wave; jt < NTILES; jt += 2 * WAVES_PER_BLOCK) {
        const int j0 = jt * TILE;
        const int j1 = (jt + WAVES_PER_BLOCK) * TILE;
        const float* brow0 = x + (size_t)(j0 + lm) * KDIM + 2 * half;
        const float* brow1 = x + (size_t)(j1 + lm) * KDIM + 2 * half;

        // Per-tile loss-mining scalars (issue early, consumed after WMMAs).
        float sqj0 = sq[j0 + lm];
        float sqj1 = sq[j1 + lm];
        int   tj0  = (int)tgt[j0 + lm];
        int   tj1  = (int)tgt[j1 + lm];

        // Preload both B tiles (compiler rolls these into clause groups of
        // in-flight b64 loads feeding the WMMA stream).
        v2f b0[KDIM / 4];
        v2f b1[KDIM / 4];
#pragma unroll
        for (int kk = 0; kk < KDIM / 4; ++kk) {
            b0[kk] = *reinterpret_cast<const v2f*>(brow0 + 4 * kk);
            b1[kk] = *reinterpret_cast<const v2f*>(brow1 + 4 * kk);
        }

        // Four independent accumulation chains (2 per tile).
        v8f acc0a = {}, acc0b = {};
        v8f acc1a = {}, acc1b = {};
#pragma unroll
        for (int kk = 0; kk < KDIM / 4; kk += 2) {
            acc0a = __builtin_amdgcn_wmma_f32_16x16x4_f32(
                false, a[kk],     false, b0[kk],     (short)0, acc0a, false, false);
            acc1a = __builtin_amdgcn_wmma_f32_16x16x4_f32(
                false, a[kk],     false, b1[kk],     (short)0, acc1a, false, false);
            acc0b = __builtin_amdgcn_wmma_f32_16x16x4_f32(
                false, a[kk + 1], false, b0[kk + 1], (short)0, acc0b, false, false);
            acc1b = __builtin_amdgcn_wmma_f32_16x16x4_f32(
                false, a[kk + 1], false, b1[kk + 1], (short)0, acc1b, false, false);
        }

#pragma unroll
        for (int v = 0; v < 8; ++v) {
            float c0  = acc0a[v] + acc0b[v];
            float c1  = acc1a[v] + acc1b[v];
            float d20 = sqi[v] + sqj0 - 2.0f * c0;
            float d21 = sqi[v] + sqj1 - 2.0f * c1;
            float d0  = sqrtf(fmaxf(d20, EPSF));
            float d1  = sqrtf(fmaxf(d21, EPSF));
            bool same0 = (ti[v] == tj0);
            bool same1 = (ti[v] == tj1);
            // branch-free selects: EXEC stays all-ones for the WMMAs above
            ap[v] = same0 ? fmaxf(ap[v], d0) : ap[v];
            an[v] = same0 ? an[v] : fminf(an[v], d0);
            ap[v] = same1 ? fmaxf(ap[v], d1) : ap[v];
            an[v] = same1 ? an[v] : fminf(an[v], d1);
        }
    }

    // Reduce across the 16 lanes of each half (each lane owns columns
    // n === lm (mod 16); rows m = v + 8*half).
#pragma unroll
    for (int v = 0; v < 8; ++v) {
#pragma unroll
        for (int off = 1; off < 16; off <<= 1) {
            ap[v] = fmaxf(ap[v], __shfl_xor(ap[v], off, 32));
            an[v] = fminf(an[v], __shfl_xor(an[v], off, 32));
        }
    }

    // Cross-wave combine via LDS.
    __shared__ float s_ap[WAVES_PER_BLOCK][TILE];
    __shared__ float s_an[WAVES_PER_BLOCK][TILE];
    if (lm == 0) {
#pragma unroll
        for (int v = 0; v < 8; ++v) {
            s_ap[wave][v + 8 * half] = ap[v];
            s_an[wave][v + 8 * half] = an[v];
        }
    }
    __syncthreads();

    if (threadIdx.x < TILE) {
        float A = -INFINITY, B = INFINITY;
#pragma unroll
        for (int w = 0; w < WAVES_PER_BLOCK; ++w) {
            A = fmaxf(A, s_ap[w][threadIdx.x]);
            B = fminf(B, s_an[w][threadIdx.x]);
        }
        float m = *marginp;
        rowloss[i0 + threadIdx.x] = fmaxf(A - B + m, 0.0f);
    }
}

// ---------------------------------------------------------------------------
// Kernel C: mean of per-row losses -> scalar
// ---------------------------------------------------------------------------
__global__ void reduce_kernel(const float* __restrict__ rowloss, float* __restrict__ out) {
    __shared__ float s[256];
    float acc = 0.0f;
    for (int i = threadIdx.x; i < NROWS; i += 256) acc += rowloss[i];
    s[threadIdx.x] = acc;
    __syncthreads();
    for (int stride = 128; stride > 0; stride >>= 1) {
        if ((int)threadIdx.x < stride) s[threadIdx.x] += s[threadIdx.x + stride];
        __syncthreads();
    }
    if (threadIdx.x == 0) out[0] = s[0] / (float)NROWS;
}

// ---------------------------------------------------------------------------
extern "C" void kernel_launch(void* const* d_in, const int* in_sizes, int n_in,
                              void* d_out, int out_size, void* d_ws, size_t ws_size,
                              hipStream_t stream) {
    const float*     x      = (const float*)d_in[0];
    const long long* tgt    = (const long long*)d_in[1];
    const float*     margin = (const float*)d_in[2];

    float* sq      = (float*)d_ws;           // 8192 floats
    float* rowloss = sq + NROWS;             // 8192 floats

    sqnorm_kernel<<<NROWS / 256, 256, 0, stream>>>(x, sq);
    gram_hard_kernel<<<NTILES, 32 * WAVES_PER_BLOCK, 0, stream>>>(x, tgt, sq, margin, rowloss);
    reduce_kernel<<<1, 256, 0, stream>>>(rowloss, (float*)d_out);
}